// TA_layer_72593537237169
// MI455X (gfx1250) — compile-verified
//
#include <hip/hip_runtime.h>

// Shapes from the reference
#define B_  32
#define T_  64
#define N_  300
#define D_  128
#define ND_ (N_ * D_)   // 38400 columns of the time-mix GEMM

typedef float v2f __attribute__((ext_vector_type(2)));
typedef float v8f __attribute__((ext_vector_type(8)));

// V_WMMA_F32_16X16X4_F32 : D(16x16 f32) = A(16x4 f32) * B(4x16 f32) + C
// A per-lane: lanes 0-15 -> {K=0,K=1} of row M=lane; lanes 16-31 -> {K=2,K=3}
// B per-lane: lanes 0-15 -> {K=0,K=1} of col N=lane; lanes 16-31 -> {K=2,K=3}
// C/D: VGPR v holds M=v (lanes 0-15) / M=v+8 (lanes 16-31), N=lane&15
__device__ __forceinline__ v8f wmma_f32x4(v2f a, v2f b, v8f c) {
    return __builtin_amdgcn_wmma_f32_16x16x4_f32(
        /*neg_a=*/false, a, /*neg_b=*/false, b,
        /*c_mod=*/(short)0, c, /*reuse_a=*/false, /*reuse_b=*/false);
}

// ---------------------------------------------------------------------------
// Kernel 1: one streaming pass over src computes
//   Q[b,t,n] = dot(src[b,t,n,:], wt1)        (reduce over D)
//   K[b,t,d] = sum_n wt3[n]*src[b,t,n,d]     (reduce over N)
// One block per (b,t); 8 waves; float4 (global_load_b128) per lane.
// ---------------------------------------------------------------------------
__global__ __launch_bounds__(256) void k_qk(const float* __restrict__ src,
                                            const float* __restrict__ wt1,
                                            const float* __restrict__ wt3,
                                            float* __restrict__ Q,
                                            float* __restrict__ Km) {
    const int tid  = threadIdx.x;
    const int lane = tid & 31;
    const int wave = tid >> 5;

    __shared__ float kacc[D_];
    if (tid < D_) kacc[tid] = 0.f;
    __syncthreads();

    const float4 w1   = reinterpret_cast<const float4*>(wt1)[lane];  // wt1[4*lane..]
    const float* base = src + (size_t)blockIdx.x * ND_;

    float k0 = 0.f, k1 = 0.f, k2 = 0.f, k3 = 0.f;
    for (int n = wave; n < N_; n += 8) {
        float4 v = reinterpret_cast<const float4*>(base + n * D_)[lane];
        float qp = v.x * w1.x + v.y * w1.y + v.z * w1.z + v.w * w1.w;
#pragma unroll
        for (int off = 16; off; off >>= 1) qp += __shfl_xor(qp, off, 32);
        const float w3 = wt3[n];
        k0 += w3 * v.x; k1 += w3 * v.y; k2 += w3 * v.z; k3 += w3 * v.w;
        if (lane == 0) Q[(size_t)blockIdx.x * N_ + n] = qp;
    }
    // cross-wave reduce of K partials via LDS float atomics (ds_add_f32)
    atomicAdd(&kacc[4 * lane + 0], k0);
    atomicAdd(&kacc[4 * lane + 1], k1);
    atomicAdd(&kacc[4 * lane + 2], k2);
    atomicAdd(&kacc[4 * lane + 3], k3);
    __syncthreads();
    if (tid < D_) Km[(size_t)blockIdx.x * D_ + tid] = kacc[tid];
}

// ---------------------------------------------------------------------------
// Kernel 2: per batch  M = Q[b](64x300) @ wt2(300x128)  then
//                      scores[b] = M @ K[b]^T (64x128 x 128x64)
// One block per batch, 4 waves, both GEMMs on WMMA f32 16x16x4.
// LDS strides padded to 132 (mod 64 == 4) so lane-halves hit disjoint banks.
// ---------------------------------------------------------------------------
#define MSTR 132
__global__ __launch_bounds__(128) void k_scores(const float* __restrict__ Q,
                                                const float* __restrict__ wt2,
                                                const float* __restrict__ Km,
                                                float* __restrict__ sc) {
    const int b    = blockIdx.x;
    const int tid  = threadIdx.x;
    const int lane = tid & 31;
    const int w    = tid >> 5;      // wave id = row tile
    const int lc   = lane & 15;
    const int hi   = lane >> 4;

    __shared__ __align__(16) float Mld[T_ * MSTR];
    __shared__ __align__(16) float Kld[T_ * MSTR];

    // cooperative load of K[b] (64x128) into padded LDS
    {
        const float* kb = Km + (size_t)b * T_ * D_;
        for (int e = tid * 4; e < T_ * D_; e += 128 * 4) {
            const int r = e >> 7, c = e & 127;
            float4 v = *reinterpret_cast<const float4*>(kb + e);
            *reinterpret_cast<float4*>(&Kld[r * MSTR + c]) = v;
        }
    }

    // Stage A: M = Q[b] @ wt2 ; wave w owns rows 16w..16w+15, all 8 col tiles
    v8f acc[8] = {};
    const float* qb = Q + (size_t)b * T_ * N_;
    for (int k0 = 0; k0 < N_; k0 += 4) {       // 300/4 = 75 K-steps
        const int kk = k0 + 2 * hi;
        v2f af = *reinterpret_cast<const v2f*>(qb + (size_t)(16 * w + lc) * N_ + kk);
#pragma unroll
        for (int c = 0; c < 8; ++c) {
            v2f bf;
            bf.x = wt2[(size_t)kk * D_ + 16 * c + lc];
            bf.y = wt2[(size_t)(kk + 1) * D_ + 16 * c + lc];
            acc[c] = wmma_f32x4(af, bf, acc[c]);
        }
    }
#pragma unroll
    for (int c = 0; c < 8; ++c)
#pragma unroll
        for (int v = 0; v < 8; ++v)
            Mld[(16 * w + v + 8 * hi) * MSTR + 16 * c + lc] = acc[c][v];
    __syncthreads();

    // Stage B: scores = M @ K^T ; wave w owns q rows 16w..16w+15, 4 k tiles
    v8f acc2[4] = {};
    for (int d0 = 0; d0 < D_; d0 += 4) {       // 32 K-steps
        const int dd = d0 + 2 * hi;
        v2f af = *reinterpret_cast<const v2f*>(&Mld[(16 * w + lc) * MSTR + dd]);
#pragma unroll
        for (int c = 0; c < 4; ++c) {
            v2f bf = *reinterpret_cast<const v2f*>(&Kld[(16 * c + lc) * MSTR + dd]);
            acc2[c] = wmma_f32x4(af, bf, acc2[c]);
        }
    }
    float* scb = sc + (size_t)b * T_ * T_;
#pragma unroll
    for (int c = 0; c < 4; ++c)
#pragma unroll
        for (int v = 0; v < 8; ++v)
            scb[(size_t)(16 * w + v + 8 * hi) * T_ + 16 * c + lc] = acc2[c][v];
}

// ---------------------------------------------------------------------------
// Kernel 3: softmax over the BATCH axis (axis=0), per (q,k) position.
// 4096 positions, 32-deep reduction; exps kept in registers.
// ---------------------------------------------------------------------------
__global__ __launch_bounds__(256) void k_softmax_b(const float* __restrict__ sc,
                                                   float* __restrict__ att) {
    const int idx = blockIdx.x * 256 + threadIdx.x;   // (q,k) flattened, 0..4095
    float e[B_];
    float m = -3.402823466e38f;
#pragma unroll
    for (int b = 0; b < B_; ++b) {
        float v = sc[(size_t)b * (T_ * T_) + idx];
        e[b] = v;
        m = fmaxf(m, v);
    }
    float s = 0.f;
#pragma unroll
    for (int b = 0; b < B_; ++b) {
        float x = __expf(e[b] - m);
        e[b] = x;
        s += x;
    }
    const float inv = 1.f / s;
#pragma unroll
    for (int b = 0; b < B_; ++b)
        att[(size_t)b * (T_ * T_) + idx] = e[b] * inv;
}

// ---------------------------------------------------------------------------
// Kernel 4 (dominant, 10 GFLOP): per batch
//   out[b,a,col] = src[b,a,col] + sum_t att[b,t,a] * src[b,t,col]
// = residual + att[b]^T (64x64) @ src[b] (64x38400), on WMMA f32 16x16x4.
//
// Block = (256-col chunk, batch); 4 waves, each wave owns 4 x 16-col strips
// -> 16 v8f accumulators/wave; each K-step issues 16 WMMAs from only
// 4 A-frags (reused across strips) + 4 B-frags (reused across row tiles).
// att[b] staged once per block (amortized 4x better than a 64-col tile:
// cuts att L2 re-reads from ~307MB to ~77MB). Streamed src strip is stashed
// in LDS so the residual add needs no second HBM pass.
// Bank padding: ASTR=72 (2 rows apart -> +16 banks), SSTR=268 (epilogue rows
// 8 apart -> +32 banks; store rows 2 apart -> +24 banks): conflict-free.
// ---------------------------------------------------------------------------
#define ASTR 72
#define SSTR 268
#define CB_  256                       // columns per block
__global__ __launch_bounds__(128) void k_mix(const float* __restrict__ src,
                                             const float* __restrict__ att,
                                             float* __restrict__ out) {
    const int tid  = threadIdx.x;
    const int lane = tid & 31;
    const int w    = tid >> 5;
    const int lc   = lane & 15;
    const int hi   = lane >> 4;
    const int b    = blockIdx.y;
    const int colBase = blockIdx.x * CB_;

    __shared__ __align__(16) float attL[T_ * ASTR];   // [t][a]
    __shared__ float srcL[T_ * SSTR];                 // [t][local col]

    // cooperative att[b] (64x64) load into padded LDS
    {
        const float* ab = att + (size_t)b * T_ * T_;
        for (int e = tid * 4; e < T_ * T_; e += 128 * 4) {
            const int t = e >> 6, a = e & 63;
            float4 v = *reinterpret_cast<const float4*>(ab + e);
            *reinterpret_cast<float4*>(&attL[t * ASTR + a]) = v;
        }
    }
    __syncthreads();

    const int    wcol = 64 * w;               // wave's 64-col slice within block
    const float* sb   = src + (size_t)b * T_ * ND_;

    v8f acc[4][4] = {};                       // [row tile r][col strip s]
    for (int t0 = 0; t0 < T_; t0 += 4) {      // 16 K-steps over t
        const int r0 = t0 + 2 * hi;

        // prefetch the next K-step's rows of the src stream (global_prefetch_b8)
        if (t0 + 4 < T_)
            __builtin_prefetch((const void*)(sb + (size_t)(r0 + 4) * ND_ + colBase + wcol + lc), 0, 3);

        v2f bf[4];
#pragma unroll
        for (int s = 0; s < 4; ++s) {         // load 4 B-frags (own strips)
            const int lcol = wcol + 16 * s + lc;
            const float bx = sb[(size_t)r0 * ND_ + colBase + lcol];
            const float by = sb[(size_t)(r0 + 1) * ND_ + colBase + lcol];
            srcL[r0 * SSTR + lcol]       = bx;   // stash for residual (wave-private)
            srcL[(r0 + 1) * SSTR + lcol] = by;
            bf[s].x = bx; bf[s].y = by;
        }
        v2f af[4];
#pragma unroll
        for (int r = 0; r < 4; ++r) {         // load 4 A-frags (shared across strips)
            af[r].x = attL[r0 * ASTR + 16 * r + lc];
            af[r].y = attL[(r0 + 1) * ASTR + 16 * r + lc];
        }
#pragma unroll
        for (int r = 0; r < 4; ++r)
#pragma unroll
            for (int s = 0; s < 4; ++s)
                acc[r][s] = wmma_f32x4(af[r], bf[s], acc[r][s]);
    }

    // epilogue: out = residual (from LDS, written by this wave) + accumulators
    float* ob = out + (size_t)b * T_ * ND_;
#pragma unroll
    for (int r = 0; r < 4; ++r)
#pragma unroll
        for (int s = 0; s < 4; ++s)
#pragma unroll
            for (int v = 0; v < 8; ++v) {
                const int a    = 16 * r + v + 8 * hi;
                const int lcol = wcol + 16 * s + lc;
                ob[(size_t)a * ND_ + colBase + lcol] = srcL[a * SSTR + lcol] + acc[r][s][v];
            }
}

// ---------------------------------------------------------------------------
extern "C" void kernel_launch(void* const* d_in, const int* in_sizes, int n_in,
                              void* d_out, int out_size, void* d_ws, size_t ws_size,
                              hipStream_t stream) {
    const float* src = (const float*)d_in[0];   // [B,T,N,D]
    const float* wt1 = (const float*)d_in[1];   // [D]
    const float* wt2 = (const float*)d_in[2];   // [N,D]
    const float* wt3 = (const float*)d_in[3];   // [N]
    float* out = (float*)d_out;                 // [B,T,N,D]

    // workspace layout (~4.6 MB of fp32)
    float* qs = (float*)d_ws;                         // Q      [B,T,N]
    float* km = qs + (size_t)B_ * T_ * N_;            // K      [B,T,D]
    float* sc = km + (size_t)B_ * T_ * D_;            // scores [B,T,T]
    float* at = sc + (size_t)B_ * T_ * T_;            // att    [B,T,T]

    k_qk<<<dim3(B_ * T_), dim3(256), 0, stream>>>(src, wt1, wt3, qs, km);
    k_scores<<<dim3(B_), dim3(128), 0, stream>>>(qs, wt2, km, sc);
    k_softmax_b<<<dim3((T_ * T_) / 256), dim3(256), 0, stream>>>(sc, at);
    k_mix<<<dim3(ND_ / CB_, B_), dim3(128), 0, stream>>>(src, at, out);
}